// gnn_backbone_64501818851773
// MI455X (gfx1250) — compile-verified
//
#include <hip/hip_runtime.h>
#include <cstdint>
#include <cstddef>

typedef __attribute__((ext_vector_type(2))) float v2f;
typedef __attribute__((ext_vector_type(8))) float v8f;

#define NNODES 100000
#define NEDGES 1000000
#define DFEAT  64
#define ND     (NNODES * DFEAT)
#define NEG_SLOPE 0.01f

// ---------------------------------------------------------------------------
// zero a float buffer
// ---------------------------------------------------------------------------
__global__ void zero_f32(float* __restrict__ p, int n) {
    int i = blockIdx.x * blockDim.x + threadIdx.x;
    if (i < n) p[i] = 0.0f;
}

// ---------------------------------------------------------------------------
// Fused triple GEMM via V_WMMA_F32_16X16X4_F32:
//   a_out = x @ W1 + b1
//   b_out = x @ W2
//   y_out = x @ W3 + b3     (y_out doubles as the segment-sum accumulator)
// Block = 128 threads (4 waves). Block handles one 16-row M-tile.
// Wave w handles N-tile w (16 cols) for all three weight matrices.
// ---------------------------------------------------------------------------
__global__ __launch_bounds__(128)
void gemm3_wmma(const float* __restrict__ x,
                const float* __restrict__ W1, const float* __restrict__ b1,
                const float* __restrict__ W2,
                const float* __restrict__ W3, const float* __restrict__ b3,
                float* __restrict__ a_out, float* __restrict__ b_out,
                float* __restrict__ y_out)
{
    // X tile stored TRANSPOSED in LDS: xs[k*16 + m]  (conflict-free A-frag loads)
    __shared__ float xs[DFEAT * 16];

    const int row0 = blockIdx.x * 16;
    const int tid  = threadIdx.x;

    // Cooperative load of 16x64 tile (1024 floats, 8 per thread) + transpose
    {
        const float4* xg = (const float4*)(x + (size_t)row0 * DFEAT);
        float4 v0 = xg[tid * 2 + 0];
        float4 v1 = xg[tid * 2 + 1];
        int idx = tid * 8;
        int r = idx >> 6;   // row in tile (0..15)
        int c = idx & 63;   // starting col
        xs[(c + 0) * 16 + r] = v0.x;  xs[(c + 1) * 16 + r] = v0.y;
        xs[(c + 2) * 16 + r] = v0.z;  xs[(c + 3) * 16 + r] = v0.w;
        xs[(c + 4) * 16 + r] = v1.x;  xs[(c + 5) * 16 + r] = v1.y;
        xs[(c + 6) * 16 + r] = v1.z;  xs[(c + 7) * 16 + r] = v1.w;
    }
    __syncthreads();

    const int lane = tid & 31;
    const int wave = tid >> 5;
    const int col  = lane & 15;   // N column (B/C/D) and M row (A) within tile
    const int hi   = lane >> 4;   // half-wave select
    const int n0   = wave * 16;   // N-tile base

    v8f acc1 = {}; v8f acc2 = {}; v8f acc3 = {};

    // A 16x4 frag layout: lanes 0-15 hold K=kk,kk+1 ; lanes 16-31 hold K=kk+2,kk+3
    // B 4x16 frag layout: VGPR0 = rows K={kk|kk+2}, VGPR1 = rows K={kk+1|kk+3}
#pragma unroll
    for (int kk = 0; kk < DFEAT; kk += 4) {
        const int kb = kk + 2 * hi;
        v2f af;
        af.x = xs[(kb + 0) * 16 + col];
        af.y = xs[(kb + 1) * 16 + col];

        v2f bf1, bf2, bf3;
        bf1.x = W1[(kb + 0) * DFEAT + n0 + col];
        bf1.y = W1[(kb + 1) * DFEAT + n0 + col];
        bf2.x = W2[(kb + 0) * DFEAT + n0 + col];
        bf2.y = W2[(kb + 1) * DFEAT + n0 + col];
        bf3.x = W3[(kb + 0) * DFEAT + n0 + col];
        bf3.y = W3[(kb + 1) * DFEAT + n0 + col];

        acc1 = __builtin_amdgcn_wmma_f32_16x16x4_f32(false, af, false, bf1,
                                                     (short)0, acc1, false, false);
        acc2 = __builtin_amdgcn_wmma_f32_16x16x4_f32(false, af, false, bf2,
                                                     (short)0, acc2, false, false);
        acc3 = __builtin_amdgcn_wmma_f32_16x16x4_f32(false, af, false, bf3,
                                                     (short)0, acc3, false, false);
    }

    // C/D layout: VGPR v, lanes 0-15 -> row (v), lanes 16-31 -> row (v+8); col = lane%16
    const float bias1 = b1[n0 + col];
    const float bias3 = b3[n0 + col];
#pragma unroll
    for (int v = 0; v < 8; ++v) {
        const int    r = row0 + v + 8 * hi;
        const size_t o = (size_t)r * DFEAT + n0 + col;
        a_out[o] = acc1[v] + bias1;
        b_out[o] = acc2[v];
        y_out[o] = acc3[v] + bias3;
    }
}

// ---------------------------------------------------------------------------
// Edge scatter: y[dst] += w * a[src]  (64 floats), deg[dst] += w
// 16 lanes per edge, each lane a float4 slab -> contiguous 256B gather/scatter.
// ---------------------------------------------------------------------------
__global__ __launch_bounds__(256)
void edge_scatter(const long long* __restrict__ ei,   // [2, E] int64
                  const float* __restrict__ ew,
                  const float* __restrict__ a_buf,
                  float* __restrict__ y_acc,
                  float* __restrict__ deg)
{
    const long long t = (long long)blockIdx.x * blockDim.x + threadIdx.x;
    const long long e = t >> 4;
    if (e >= NEDGES) return;
    const int l = (int)(t & 15);

    const long long src = ei[e];
    const long long dst = ei[NEDGES + e];
    const float     w   = ew[e];

    const float4 av = *(const float4*)(a_buf + src * DFEAT + l * 4);
    float* yp = y_acc + dst * DFEAT + l * 4;
    atomicAdd(yp + 0, w * av.x);
    atomicAdd(yp + 1, w * av.y);
    atomicAdd(yp + 2, w * av.z);
    atomicAdd(yp + 3, w * av.w);
    if (l == 0) atomicAdd(deg + dst, w);
}

// ---------------------------------------------------------------------------
// Finalize: x_next = leaky_relu(y - deg * b)
// ---------------------------------------------------------------------------
__global__ __launch_bounds__(256)
void finalize_layer(const float* __restrict__ y_acc,
                    const float* __restrict__ b_buf,
                    const float* __restrict__ deg,
                    float* __restrict__ x_next)
{
    const int i = blockIdx.x * blockDim.x + threadIdx.x;
    if (i >= ND) return;
    const float d = deg[i >> 6];
    const float v = y_acc[i] - d * b_buf[i];
    x_next[i] = (v >= 0.0f) ? v : NEG_SLOPE * v;
}

// ---------------------------------------------------------------------------
extern "C" void kernel_launch(void* const* d_in, const int* in_sizes, int n_in,
                              void* d_out, int out_size, void* d_ws, size_t ws_size,
                              hipStream_t stream)
{
    const float*     x_in = (const float*)d_in[0];
    const long long* ei   = (const long long*)d_in[1];   // int64 edge_index [2,E]
    const float*     ew   = (const float*)d_in[2];
    const float*     W1a  = (const float*)d_in[3];
    const float*     b1a  = (const float*)d_in[4];
    const float*     W2a  = (const float*)d_in[5];
    const float*     W3a  = (const float*)d_in[6];
    const float*     b3a  = (const float*)d_in[7];

    float* ws    = (float*)d_ws;
    float* a_buf = ws;                       // ND floats
    float* b_buf = ws + (size_t)ND;          // ND floats
    float* x_mid = ws + 2 * (size_t)ND;      // ND floats
    float* deg   = ws + 3 * (size_t)ND;      // NNODES floats
    float* y_acc = (float*)d_out;            // ND floats: per-layer accumulator

    const int mtiles      = NNODES / 16;                       // 6250 (exact)
    const int edge_blocks = (int)(((long long)NEDGES * 16 + 255) / 256);
    const int nd_blocks   = (ND + 255) / 256;
    const int deg_blocks  = (NNODES + 255) / 256;

    for (int layer = 0; layer < 3; ++layer) {
        const float* xin  = (layer == 0) ? x_in : x_mid;
        float*       xout = (layer == 2) ? (float*)d_out : x_mid;
        const float* W1 = W1a + (size_t)layer * DFEAT * DFEAT;
        const float* b1 = b1a + (size_t)layer * DFEAT;
        const float* W2 = W2a + (size_t)layer * DFEAT * DFEAT;
        const float* W3 = W3a + (size_t)layer * DFEAT * DFEAT;
        const float* b3 = b3a + (size_t)layer * DFEAT;

        zero_f32<<<deg_blocks, 256, 0, stream>>>(deg, NNODES);
        gemm3_wmma<<<mtiles, 128, 0, stream>>>(xin, W1, b1, W2, W3, b3,
                                               a_buf, b_buf, y_acc);
        edge_scatter<<<edge_blocks, 256, 0, stream>>>(ei, ew, a_buf, y_acc, deg);
        finalize_layer<<<nd_blocks, 256, 0, stream>>>(y_acc, b_buf, deg, xout);
    }
}